// FFT_features_11622181503298
// MI455X (gfx1250) — compile-verified
//
#include <hip/hip_runtime.h>
#include <math.h>

typedef float v2f __attribute__((ext_vector_type(2)));
typedef float v4f __attribute__((ext_vector_type(4)));
typedef float v8f __attribute__((ext_vector_type(8)));

#define WMMA4(a, b, c) \
  __builtin_amdgcn_wmma_f32_16x16x4_f32(false, (a), false, (b), (short)0, (c), false, false)

// Folded-weight records in WMMA A-operand layout (64 floats each: [lane][2]).
// rec 0        : stage A (16x4,  1 mtile, 1 chunk)
// rec 1..10    : stage B (32x20, 2 mtiles, 5 chunks)  [FFT folded, bias col @k=16]
// rec 11..26   : stage C (32x32, 2 mtiles, 8 chunks)  [IFFT real/imag]
// rec 27..46   : stage D (64x20, 4 mtiles, 5 chunks)  [bias col @k=16]
#define NREC 47

static constexpr float TWO_PI_OVER_16 = 0.392699081698724154808f;

__global__ __launch_bounds__(256) void fft_features_fused(
    const float* __restrict__ img,
    const float* __restrict__ w_in,
    const float* __restrict__ g1, const float* __restrict__ b1,
    const float* __restrict__ m1, const float* __restrict__ v1,
    const float* __restrict__ w_mid,
    const float* __restrict__ g2, const float* __restrict__ b2,
    const float* __restrict__ m2, const float* __restrict__ v2,
    const float* __restrict__ w_out,
    const float* __restrict__ g3, const float* __restrict__ b3,
    const float* __restrict__ m3, const float* __restrict__ v3,
    float* __restrict__ out)
{
    __shared__ __align__(16) float wgt[NREC * 64];      // 12032 B
    __shared__ __align__(16) float smem[8 * 1216];      // 38912 B (per-wave tiles)

    const int tid = threadIdx.x;

    // ---------- Fold BN + DFT/IDFT into WMMA-layout weights (redundant per WG, tiny) ----------
    for (int i = tid; i < NREC * 64; i += 256) {
        const int rec  = i >> 6;
        const int ln   = (i >> 1) & 31;
        const int v    = i & 1;
        const int mloc = ln & 15;
        const int koff = (ln < 16) ? 0 : 2;
        float val = 0.0f;
        if (rec == 0) {                                   // stage A: 16x4
            const int k = koff + v;
            const int m = mloc;
            const float s = g1[m] * rsqrtf(v1[m] + 1e-5f);
            val = (k < 3) ? s * w_in[m * 3 + k] : (b1[m] - m1[m] * s);
        } else if (rec < 11) {                            // stage B: 32x20
            const int r = rec - 1, mt = r / 5, ch = r % 5;
            const int k = ch * 4 + koff + v;
            const int m = mt * 16 + mloc;
            const float s = g2[m] * rsqrtf(v2[m] + 1e-5f);
            if (k < 16) {
                float acc = 0.0f;
                for (int j = 0; j < 32; ++j) {            // w_mid @ [Re F; Im F]
                    const int jj = (j < 16) ? j : (j - 16);
                    const float ang = (float)((jj * k) & 15) * TWO_PI_OVER_16;
                    const float mf = (j < 16) ? cosf(ang) : -sinf(ang);
                    acc += w_mid[m * 32 + j] * mf;
                }
                val = s * acc;
            } else if (k == 16) {
                val = b2[m] - m2[m] * s;
            }
        } else if (rec < 27) {                            // stage C: 32x32 IFFT
            const int r = rec - 11, mt = r / 8, ch = r % 8;
            const int k = ch * 4 + koff + v;
            const int m = mt * 16 + mloc;
            const int mr = m & 15, kk = k & 15;
            const float ang = (float)((mr * kk) & 15) * TWO_PI_OVER_16;
            const float cr = cosf(ang) * 0.0625f;
            const float si = sinf(ang) * 0.0625f;
            if (m < 16) val = (k < 16) ? cr : -si;        // z_real = Gr*a - Gi*b
            else        val = (k < 16) ? si :  cr;        // z_imag = Gi*a + Gr*b
        } else {                                          // stage D: 64x20
            const int r = rec - 27, mt = r / 5, ch = r % 5;
            const int k = ch * 4 + koff + v;
            const int m = mt * 16 + mloc;
            const float s = g3[m] * rsqrtf(v3[m] + 1e-5f);
            if (k < 16)       val = s * w_out[m * 16 + k];
            else if (k == 16) val = b3[m] - m3[m] * s;
        }
        wgt[i] = val;
    }
    __syncthreads();

    const int lane = tid & 31;
    const int wv   = tid >> 5;
    const int n    = lane & 15;        // pixel within tile
    const int h    = lane >> 4;        // half-wave

    // ---------- Load A operands (resident in VGPRs) ----------
    v2f wA = *(const v2f*)(wgt + lane * 2);
    v2f wB[2][5], wC[2][8], wD[4][5];
#pragma unroll
    for (int mt = 0; mt < 2; ++mt)
#pragma unroll
        for (int c = 0; c < 5; ++c)
            wB[mt][c] = *(const v2f*)(wgt + (1 + mt * 5 + c) * 64 + lane * 2);
#pragma unroll
    for (int mt = 0; mt < 2; ++mt)
#pragma unroll
        for (int c = 0; c < 8; ++c)
            wC[mt][c] = *(const v2f*)(wgt + (11 + mt * 8 + c) * 64 + lane * 2);
#pragma unroll
    for (int mt = 0; mt < 4; ++mt)
#pragma unroll
        for (int c = 0; c < 5; ++c)
            wD[mt][c] = *(const v2f*)(wgt + (27 + mt * 5 + c) * 64 + lane * 2);

    float* my    = smem + wv * 1216;
    float* ldsY1 = my;          // 16 px, stride 20 floats
    float* ldsY2 = my + 320;    // 16 px, stride 36 floats
    float* ldsMG = my + 896;    // 16 px, stride 20 floats

    constexpr int HW = 512 * 1024;
    constexpr int NT = 4 * HW / 16;    // 131072 tiles
    const int stride = (int)(gridDim.x << 3);
    const v8f zro = {0, 0, 0, 0, 0, 0, 0, 0};

    for (int t = blockIdx.x * 8 + wv; t < NT; t += stride) {
        const int b   = t >> 15;               // 32768 tiles per batch
        const int hw0 = (t & 32767) << 4;

        // ---- stage A: 16x4 @ 4x16 (ch3 = constant-1 bias row) ----
        const float* ip = img + (size_t)(b * 3) * HW + hw0 + n;
        const float x0 = __builtin_nontemporal_load(ip + (size_t)(2 * h) * HW);
        const float xa = __builtin_nontemporal_load(ip + (size_t)(h ? 0 : 1) * HW);
        v2f bA = {x0, h ? 1.0f : xa};
        v8f y1 = WMMA4(wA, bA, zro);
#pragma unroll
        for (int r = 0; r < 8; ++r) y1[r] = fmaxf(y1[r], 0.0f);
        {
            v4f lo = {y1[0], y1[1], y1[2], y1[3]};
            v4f hi = {y1[4], y1[5], y1[6], y1[7]};
            *(v4f*)(ldsY1 + n * 20 + 8 * h)     = lo;
            *(v4f*)(ldsY1 + n * 20 + 8 * h + 4) = hi;
        }

        // ---- stage B: (32x20) @ (20x16), FFT folded ----
        v8f a0 = zro, a1 = zro;
#pragma unroll
        for (int c = 0; c < 4; ++c) {
            v2f bv = *(const v2f*)(ldsY1 + n * 20 + c * 4 + 2 * h);
            a0 = WMMA4(wB[0][c], bv, a0);
            a1 = WMMA4(wB[1][c], bv, a1);
        }
        {
            v2f bv = {h ? 0.0f : 1.0f, 0.0f};  // bias row k=16
            a0 = WMMA4(wB[0][4], bv, a0);
            a1 = WMMA4(wB[1][4], bv, a1);
        }
#pragma unroll
        for (int r = 0; r < 8; ++r) { a0[r] = fmaxf(a0[r], 0.0f); a1[r] = fmaxf(a1[r], 0.0f); }
        {
            v4f p0 = {a0[0], a0[1], a0[2], a0[3]};
            v4f p1 = {a0[4], a0[5], a0[6], a0[7]};
            v4f p2 = {a1[0], a1[1], a1[2], a1[3]};
            v4f p3 = {a1[4], a1[5], a1[6], a1[7]};
            *(v4f*)(ldsY2 + n * 36 + 8 * h)          = p0;
            *(v4f*)(ldsY2 + n * 36 + 8 * h + 4)      = p1;
            *(v4f*)(ldsY2 + n * 36 + 16 + 8 * h)     = p2;
            *(v4f*)(ldsY2 + n * 36 + 16 + 8 * h + 4) = p3;
        }

        // ---- stage C: IFFT (32x32) @ (32x16), then |z| ----
        v8f zr = zro, zi = zro;
#pragma unroll
        for (int c = 0; c < 8; ++c) {
            v2f bv = *(const v2f*)(ldsY2 + n * 36 + c * 4 + 2 * h);
            zr = WMMA4(wC[0][c], bv, zr);
            zi = WMMA4(wC[1][c], bv, zi);
        }
        v8f mg;
#pragma unroll
        for (int r = 0; r < 8; ++r) mg[r] = sqrtf(zr[r] * zr[r] + zi[r] * zi[r]);
        {
            v4f lo = {mg[0], mg[1], mg[2], mg[3]};
            v4f hi = {mg[4], mg[5], mg[6], mg[7]};
            *(v4f*)(ldsMG + n * 20 + 8 * h)     = lo;
            *(v4f*)(ldsMG + n * 20 + 8 * h + 4) = hi;
        }

        // ---- stage D: (64x20) @ (20x16) ----
        v8f o0 = zro, o1 = zro, o2 = zro, o3 = zro;
#pragma unroll
        for (int c = 0; c < 4; ++c) {
            v2f bv = *(const v2f*)(ldsMG + n * 20 + c * 4 + 2 * h);
            o0 = WMMA4(wD[0][c], bv, o0);
            o1 = WMMA4(wD[1][c], bv, o1);
            o2 = WMMA4(wD[2][c], bv, o2);
            o3 = WMMA4(wD[3][c], bv, o3);
        }
        {
            v2f bv = {h ? 0.0f : 1.0f, 0.0f};
            o0 = WMMA4(wD[0][4], bv, o0);
            o1 = WMMA4(wD[1][4], bv, o1);
            o2 = WMMA4(wD[2][4], bv, o2);
            o3 = WMMA4(wD[3][4], bv, o3);
        }

        // ---- streaming store [B,64,H,W] ----
        float* op = out + (size_t)(b * 64 + 8 * h) * HW + hw0 + n;
#pragma unroll
        for (int r = 0; r < 8; ++r) {
            __builtin_nontemporal_store(fmaxf(o0[r], 0.0f), op + (size_t)(r)      * HW);
            __builtin_nontemporal_store(fmaxf(o1[r], 0.0f), op + (size_t)(r + 16) * HW);
            __builtin_nontemporal_store(fmaxf(o2[r], 0.0f), op + (size_t)(r + 32) * HW);
            __builtin_nontemporal_store(fmaxf(o3[r], 0.0f), op + (size_t)(r + 48) * HW);
        }
    }
}

extern "C" void kernel_launch(void* const* d_in, const int* in_sizes, int n_in,
                              void* d_out, int out_size, void* d_ws, size_t ws_size,
                              hipStream_t stream) {
    (void)in_sizes; (void)n_in; (void)d_ws; (void)ws_size; (void)out_size;
    const float* img   = (const float*)d_in[0];
    const float* w_in  = (const float*)d_in[1];
    const float* g1    = (const float*)d_in[2];
    const float* b1    = (const float*)d_in[3];
    const float* m1    = (const float*)d_in[4];
    const float* v1    = (const float*)d_in[5];
    const float* w_mid = (const float*)d_in[6];
    const float* g2    = (const float*)d_in[7];
    const float* b2    = (const float*)d_in[8];
    const float* m2    = (const float*)d_in[9];
    const float* v2    = (const float*)d_in[10];
    const float* w_out = (const float*)d_in[11];
    const float* g3    = (const float*)d_in[12];
    const float* b3    = (const float*)d_in[13];
    const float* m3    = (const float*)d_in[14];
    const float* v3    = (const float*)d_in[15];
    float* out = (float*)d_out;

    hipLaunchKernelGGL(fft_features_fused, dim3(2048), dim3(256), 0, stream,
                       img, w_in, g1, b1, m1, v1, w_mid, g2, b2, m2, v2,
                       w_out, g3, b3, m3, v3, out);
}